// Attention_90254442758625
// MI455X (gfx1250) — compile-verified
//
#include <hip/hip_runtime.h>

// Paged-attention GQA decode for MI455X (gfx1250, wave32).
// Bandwidth-bound (~537MB K/V @ 23.3TB/s => ~23us floor); f32 WMMA 16x16x4
// keeps reference precision since matrix throughput is not the bottleneck.
// K/V tiles are streamed memory->LDS with CDNA5 async-to-LDS ops (ASYNCcnt),
// double-buffered so each wave always has a full tile in flight. The stream
// is single-use and larger than L2 (512MB > 192MB), so loads carry TH=NT.

#define BATCH_   16
#define HQ_      32
#define HKV_     8
#define HD_      128
#define BLKSZ_   256
#define MAXBLK_  16
#define NREP_    4          // HQ_/HKV_
#define SPLIT_   32         // independent wave-partitions per (b, kv_head)
#define TILE_    16         // tokens per WMMA tile
#define KST_     132        // padded LDS row stride (floats) -> conflict-free b64 frag reads

typedef __attribute__((ext_vector_type(2))) float v2f;
typedef __attribute__((ext_vector_type(8))) float v8f;

// One async row: 32 lanes x b128 = one 512B K/V row, global -> LDS, no VGPRs.
// Non-temporal: K/V bytes are consumed exactly once per launch.
__device__ __forceinline__ void async_row_b128(unsigned lds_byte, const float* gsrc) {
  asm volatile("global_load_async_to_lds_b128 %0, %1, off th:TH_LOAD_NT"
               :: "v"(lds_byte), "v"((unsigned long long)(uintptr_t)gsrc)
               : "memory");
}

// ---------------------------------------------------------------------------
// Kernel 1: one wave32 workgroup per (b, g, slot). Flash-decode over tiles
// t = slot, slot+SPLIT, ... ; writes unnormalized partial O / m / l.
// ---------------------------------------------------------------------------
__global__ __launch_bounds__(32) void pa_decode_partial(
    const float* __restrict__ q,     // [B, HQ, HD]
    const float* __restrict__ knew,  // [B, HKV, HD]
    const float* __restrict__ vnew,  // [B, HKV, HD]
    const float* __restrict__ kc,    // [NB, BLK, HKV, HD]
    const float* __restrict__ vc,
    const int*   __restrict__ ctx,   // [B]
    const int*   __restrict__ btab,  // [B, MAXBLK]
    float* __restrict__ ws_o,        // [B][HKV][SPLIT][8][16][4]
    float* __restrict__ ws_ml)       // [B][HKV][SPLIT][8] = m[4], l[4]
{
  const int lane = threadIdx.x & 31;
  const int slot = blockIdx.x;
  const int g    = blockIdx.y;
  const int b    = blockIdx.z;

  const int L = ctx[b];
  const int T = (L + TILE_ - 1) / TILE_;

  __shared__ float lK[2][TILE_ * KST_];   // double-buffered K tile
  __shared__ float lV[2][TILE_ * KST_];   // double-buffered V tile
  __shared__ float lP[TILE_ * 16];

  const int arow  = lane & 15;         // M (or N) index owned by this lane
  const int apair = (lane >> 4) * 2;   // K-dim pair within a 4-chunk (ISA 7.12.2)

  const v8f vzero = {};

  // LDS byte offsets (wave-relative; low 32 bits of the generic pointer)
  const unsigned kbase0 = (unsigned)(uintptr_t)&lK[0][0] + lane * 16;
  const unsigned kbase1 = (unsigned)(uintptr_t)&lK[1][0] + lane * 16;
  const unsigned vbase0 = (unsigned)(uintptr_t)&lV[0][0] + lane * 16;
  const unsigned vbase1 = (unsigned)(uintptr_t)&lV[1][0] + lane * 16;

  const float* knewrow = knew + ((size_t)b * HKV_ + g) * HD_;
  const float* vnewrow = vnew + ((size_t)b * HKV_ + g) * HD_;

  // ---- Q A-fragments (16x4 f32 per chunk, 32 chunks cover d=128), kept in
  // VGPRs for the whole sweep. Rows >= NREP_ are zeroed (unused GQA rows).
  const float zm = (arow < NREP_) ? 1.0f : 0.0f;
  const float* qp = q + ((size_t)b * HQ_ + g * NREP_ + (arow < NREP_ ? arow : 0)) * HD_;
  v2f qf[32];
#pragma unroll
  for (int c = 0; c < 32; ++c) {
    v2f t;
    t[0] = qp[4 * c + apair]     * zm;
    t[1] = qp[4 * c + apair + 1] * zm;
    qf[c] = t;
  }

  v8f acc[8];                       // O accumulator: 8 d-tiles of 16x16 f32
#pragma unroll
  for (int i = 0; i < 8; ++i) acc[i] = vzero;

  float mh[NREP_], lh[NREP_];
#pragma unroll
  for (int h = 0; h < NREP_; ++h) { mh[h] = -1.0e30f; lh[h] = 0.0f; }

  // scores kept in base-2 domain: s^ = s * (1/sqrt(128)) * log2(e)
  const float sc = 0.088388347648318447f * 1.4426950408889634f;

  // async-issue one 16-token tile (32 x b128 rows => ASYNCcnt += 32)
  auto issue_tile = [&](int tt, int bufi) {
    const int blk  = btab[b * MAXBLK_ + (tt >> 4)];      // 16 tiles per cache block
    const size_t rowbase = (((size_t)blk * BLKSZ_ + (tt & 15) * TILE_) * HKV_ + g) * HD_;
    const int tok0 = tt * TILE_;
    const unsigned kb = bufi ? kbase1 : kbase0;
    const unsigned vb = bufi ? vbase1 : vbase0;
#pragma unroll
    for (int r = 0; r < TILE_; ++r) {
      // row == L-1 comes from the new token (replaces the reference's cache
      // scatter without mutating the input cache buffers)
      const float* sk = (tok0 + r == L - 1) ? knewrow
                       : (kc + rowbase + (size_t)r * HKV_ * HD_);
      async_row_b128(kb + r * (KST_ * 4), sk + lane * 4);
    }
#pragma unroll
    for (int r = 0; r < TILE_; ++r) {
      const float* sv_ = (tok0 + r == L - 1) ? vnewrow
                        : (vc + rowbase + (size_t)r * HKV_ * HD_);
      async_row_b128(vb + r * (KST_ * 4), sv_ + lane * 4);
    }
  };

  int buf = 0;
  if (slot < T) issue_tile(slot, 0);

  for (int t = slot; t < T; t += SPLIT_) {
    const int tn = t + SPLIT_;
    if (tn < T) {
      issue_tile(tn, buf ^ 1);                       // pipeline the next tile
      // GL2 warm-up hint for the tile after that
      const int tn2 = tn + SPLIT_;
      if (tn2 < T) {
        const int blkn = btab[b * MAXBLK_ + (tn2 >> 4)];
        const size_t rbn = (((size_t)blkn * BLKSZ_ + (tn2 & 15) * TILE_) * HKV_ + g) * HD_;
        __builtin_prefetch(kc + rbn + (size_t)lane * 32, 0, 1);
        __builtin_prefetch(vc + rbn + (size_t)lane * 32, 0, 1);
      }
      // async loads complete in order: <=32 outstanding => current tile ready
      asm volatile("s_wait_asynccnt 0x20" ::: "memory");
    } else {
      asm volatile("s_wait_asynccnt 0x0" ::: "memory");
    }

    const float* curK = &lK[buf][0];
    const float* curV = &lV[buf][0];
    const int tok0 = t * TILE_;

    // ---- S = Q * K^T  (A = Q 16x4, B = K^T 4x16), two interleaved C chains
    v8f s0 = vzero, s1 = vzero;
#pragma unroll
    for (int c = 0; c < 32; c += 2) {
      v2f bk0 = *reinterpret_cast<const v2f*>(&curK[arow * KST_ + 4 * c + apair]);
      v2f bk1 = *reinterpret_cast<const v2f*>(&curK[arow * KST_ + 4 * (c + 1) + apair]);
      s0 = __builtin_amdgcn_wmma_f32_16x16x4_f32(false, qf[c],     false, bk0, (short)0, s0, false, false);
      s1 = __builtin_amdgcn_wmma_f32_16x16x4_f32(false, qf[c + 1], false, bk1, (short)0, s1, false, false);
    }

    // ---- online softmax (rows 0..3 valid; C layout: VGPR h = row h, N = lane)
    const bool tvalid = (tok0 + arow) < L;
    float sv[NREP_];
#pragma unroll
    for (int h = 0; h < NREP_; ++h)
      sv[h] = tvalid ? (s0[h] + s1[h]) * sc : -1.0e30f;

    float al[NREP_];
#pragma unroll
    for (int h = 0; h < NREP_; ++h) {
      float v = sv[h];
      v = fmaxf(v, __shfl_xor(v, 1, 32));
      v = fmaxf(v, __shfl_xor(v, 2, 32));
      v = fmaxf(v, __shfl_xor(v, 4, 32));
      v = fmaxf(v, __shfl_xor(v, 8, 32));
      v = __shfl(v, 0, 32);                       // broadcast tile max
      const float mn = fmaxf(mh[h], v);
      al[h] = exp2f(mh[h] - mn);
      mh[h] = mn;
    }

    float p[NREP_];
#pragma unroll
    for (int h = 0; h < NREP_; ++h) {
      p[h] = exp2f(sv[h] - mh[h]);
      float v = p[h];
      v += __shfl_xor(v, 1, 32);
      v += __shfl_xor(v, 2, 32);
      v += __shfl_xor(v, 4, 32);
      v += __shfl_xor(v, 8, 32);
      v = __shfl(v, 0, 32);                       // broadcast tile sum
      lh[h] = lh[h] * al[h] + v;
    }

    // rescale running O rows by alpha (rows h+8 hold don't-care data)
#pragma unroll
    for (int dt = 0; dt < 8; ++dt)
#pragma unroll
      for (int h = 0; h < NREP_; ++h)
        acc[dt][h] = acc[dt][h] * al[h];

    // ---- transpose P (C layout -> A layout) through LDS
    if (lane < 16) {
#pragma unroll
      for (int h = 0; h < NREP_; ++h) lP[h * 16 + lane] = p[h];
    }
    v2f pf[4];
#pragma unroll
    for (int c = 0; c < 4; ++c)
      pf[c] = *reinterpret_cast<const v2f*>(&lP[arow * 16 + 4 * c + apair]);

    // ---- O += P * V  (A = P 16x4, B = V 4x16 per d-tile)
#pragma unroll
    for (int dt = 0; dt < 8; ++dt) {
      const int dcol = dt * 16 + arow;
#pragma unroll
      for (int c = 0; c < 4; ++c) {
        v2f bv;
        bv[0] = curV[(4 * c + apair)     * KST_ + dcol];
        bv[1] = curV[(4 * c + apair + 1) * KST_ + dcol];
        acc[dt] = __builtin_amdgcn_wmma_f32_16x16x4_f32(false, pf[c], false, bv, (short)0, acc[dt], false, false);
      }
    }

    buf ^= 1;
  }

  // ---- write per-wave partials (written unconditionally: empty slots emit
  // m=-1e30, l=0, O=0 and contribute zero weight in the combine)
  float* wsop = ws_o + ((((size_t)b * HKV_ + g) * SPLIT_ + slot) * 8 * 16 * 4);
  if (lane < 16) {
#pragma unroll
    for (int dt = 0; dt < 8; ++dt) {
      float4 o4 = make_float4(acc[dt][0], acc[dt][1], acc[dt][2], acc[dt][3]);
      *reinterpret_cast<float4*>(wsop + (dt * 16 + lane) * 4) = o4;
    }
  }
  if (lane == 0) {
    float* mlp = ws_ml + (((size_t)b * HKV_ + g) * SPLIT_ + slot) * 8;
#pragma unroll
    for (int h = 0; h < NREP_; ++h) { mlp[h] = mh[h]; mlp[4 + h] = lh[h]; }
  }
}

// ---------------------------------------------------------------------------
// Kernel 2: merge the SPLIT_ partials per (b, kv_head), normalize, write out.
// One float4 load per slot carries all 4 GQA heads for this d.
// ---------------------------------------------------------------------------
__global__ __launch_bounds__(128) void pa_decode_combine(
    const float* __restrict__ ws_o, const float* __restrict__ ws_ml,
    float* __restrict__ out)  // [B, 1, HQ, HD]
{
  const int d = threadIdx.x;   // 0..127
  const int g = blockIdx.x;    // 0..7
  const int b = blockIdx.y;    // 0..15

  const float* mlbase = ws_ml + (((size_t)b * HKV_ + g) * SPLIT_) * 8;
  const float* obase  = ws_o  + (((size_t)b * HKV_ + g) * SPLIT_) * (8 * 16 * 4);

  float mstar[NREP_];
#pragma unroll
  for (int h = 0; h < NREP_; ++h) mstar[h] = -1.0e30f;
  for (int s = 0; s < SPLIT_; ++s) {
#pragma unroll
    for (int h = 0; h < NREP_; ++h)
      mstar[h] = fmaxf(mstar[h], mlbase[s * 8 + h]);
  }

  float osum[NREP_], lstar[NREP_];
#pragma unroll
  for (int h = 0; h < NREP_; ++h) { osum[h] = 0.0f; lstar[h] = 0.0f; }

  for (int s = 0; s < SPLIT_; ++s) {
    const float4 o4 =
        *reinterpret_cast<const float4*>(obase + (size_t)s * (8 * 16 * 4) + d * 4);
    const float oh[NREP_] = {o4.x, o4.y, o4.z, o4.w};
#pragma unroll
    for (int h = 0; h < NREP_; ++h) {
      const float w = exp2f(mlbase[s * 8 + h] - mstar[h]);
      lstar[h] += w * mlbase[s * 8 + 4 + h];
      osum[h]  += w * oh[h];
    }
  }

#pragma unroll
  for (int h = 0; h < NREP_; ++h)
    out[((size_t)b * HQ_ + g * NREP_ + h) * HD_ + d] = osum[h] / lstar[h];
}

// ---------------------------------------------------------------------------
extern "C" void kernel_launch(void* const* d_in, const int* in_sizes, int n_in,
                              void* d_out, int out_size, void* d_ws, size_t ws_size,
                              hipStream_t stream) {
  (void)in_sizes; (void)n_in; (void)out_size; (void)ws_size;
  const float* q    = (const float*)d_in[0];
  const float* knew = (const float*)d_in[1];
  const float* vnew = (const float*)d_in[2];
  const float* kc   = (const float*)d_in[3];
  const float* vc   = (const float*)d_in[4];
  const int*   ctx  = (const int*)d_in[5];
  const int*   btab = (const int*)d_in[6];
  // d_in[7] (slot_mapping) is not needed: the newest token always lands at
  // position ctx-1 of its own (disjoint) block range, so we substitute k/v
  // directly during staging instead of mutating the cache inputs.

  float* out   = (float*)d_out;
  float* ws_o  = (float*)d_ws;                                        // 8 MB
  float* ws_ml = ws_o + (size_t)BATCH_ * HKV_ * SPLIT_ * 8 * 16 * 4;  // +128 KB

  dim3 gA(SPLIT_, HKV_, BATCH_), bA(32);
  pa_decode_partial<<<gA, bA, 0, stream>>>(q, knew, vnew, kc, vc, ctx, btab, ws_o, ws_ml);

  dim3 gB(HKV_, BATCH_), bB(128);
  pa_decode_combine<<<gB, bB, 0, stream>>>(ws_o, ws_ml, out);
}